// Fume3d_58248346468419
// MI455X (gfx1250) — compile-verified
//
#include <hip/hip_runtime.h>

// Epipolar line-sum: out[p,v,u] = sum over the epipolar line l = F^T [u,v,1]
// of samples from input plane p. Plane = 256x256 f32 (256 KB) -> staged into
// LDS (rows padded to 257 dwords so bank = (y+x) mod 64 => conflict-free
// gathers in both the column- and row-dominant branches). Staging uses the
// gfx1250 async global->LDS DMA path (ASYNCcnt), the CDNA5-specific data
// movement feature for this workload.

#define IMG_H 256
#define IMG_W 256
#define LDS_STRIDE 257                 // 257 mod 64 == 1 -> bank = (y + x) % 64
#define ROWS_PER_TILE 16
#define BLOCK_THREADS 1024             // 32 wave32s -> fills one WGP
#define LDS_FLOATS (IMG_H * LDS_STRIDE)

#if defined(__HIP_DEVICE_COMPILE__) && defined(__gfx1250__) && \
    __has_builtin(__builtin_amdgcn_global_load_async_to_lds_b32)
#define USE_ASYNC_LDS 1
#else
#define USE_ASYNC_LDS 0
#endif

#if USE_ASYNC_LDS
typedef __attribute__((address_space(1))) int as1_int;
typedef __attribute__((address_space(3))) int as3_int;
#endif

__global__ __launch_bounds__(BLOCK_THREADS, 1)
void epiline_sum_kernel(const float* __restrict__ img,
                        const float* __restrict__ Fm,
                        float* __restrict__ out)
{
    __shared__ float lds[LDS_FLOATS];  // 263,168 B of the 320 KB WGP LDS

    const int plane = blockIdx.y;
    const int tile  = blockIdx.x;
    const float* __restrict__ g = img + (size_t)plane * (IMG_H * IMG_W);
    const int tid = threadIdx.x;

    // ---- Stage full plane into LDS with 257-dword padded rows ----------
    // 65536 floats / 1024 threads = 64 per thread. Lanes take consecutive
    // columns -> coalesced global side, consecutive banks on the LDS side.
    #pragma unroll
    for (int i = 0; i < (IMG_H * IMG_W) / BLOCK_THREADS; ++i) {
        const int e   = tid + i * BLOCK_THREADS;
        const int row = e >> 8;        // /256
        const int col = e & 255;
#if USE_ASYNC_LDS
        __builtin_amdgcn_global_load_async_to_lds_b32(
            (as1_int*)(g + e),
            (as3_int*)(lds + row * LDS_STRIDE + col),
            0, 0);
#else
        lds[row * LDS_STRIDE + col] = g[e];
#endif
    }
#if USE_ASYNC_LDS
#if __has_builtin(__builtin_amdgcn_s_wait_asynccnt)
    __builtin_amdgcn_s_wait_asynccnt(0);
#else
    asm volatile("s_wait_asynccnt 0" ::: "memory");
#endif
#endif
    __syncthreads();

    // ---- Line coefficients: l_i = sum_j F[j][i] * p_j, p = (u, v, 1) ----
    const float F00 = Fm[0], F01 = Fm[1], F02 = Fm[2];
    const float F10 = Fm[3], F11 = Fm[4], F12 = Fm[5];
    const float F20 = Fm[6], F21 = Fm[7], F22 = Fm[8];

    const int u  = tid & 255;                       // consecutive u per wave
    const int v0 = tile * ROWS_PER_TILE + (tid >> 8);
    const float uf = (float)u;
    float* __restrict__ outp = out + (size_t)plane * (IMG_H * IMG_W);

    #pragma unroll
    for (int k = 0; k < ROWS_PER_TILE / (BLOCK_THREADS / IMG_W); ++k) {
        const int   v  = v0 + k * (BLOCK_THREADS / IMG_W);
        const float vf = (float)v;

        const float a = fmaf(F00, uf, fmaf(F10, vf, F20));
        const float b = fmaf(F01, uf, fmaf(F11, vf, F21));
        const float c = fmaf(F02, uf, fmaf(F12, vf, F22));
        const float eps = 1e-8f;
        float acc = 0.0f;

        if (fabsf(b) >= fabsf(a)) {
            // column-dominant: y(x) = -(a*x + c) / b_safe
            const float bs  = (fabsf(b) < eps) ? eps : b;
            const float inv = -1.0f / bs;
            float xf = 0.0f;
            #pragma unroll 4
            for (int x = 0; x < IMG_W; ++x) {
                const float yy = fmaf(a, xf, c) * inv;
                const int   yi = __float2int_rn(yy);      // round-half-even
                if ((unsigned)yi < (unsigned)IMG_H)
                    acc += lds[yi * LDS_STRIDE + x];
                xf += 1.0f;
            }
        } else {
            // row-dominant: x(y) = -(b*y + c) / a_safe
            const float as  = (fabsf(a) < eps) ? eps : a;
            const float inv = -1.0f / as;
            float yf = 0.0f;
            #pragma unroll 4
            for (int y = 0; y < IMG_H; ++y) {
                const float xx = fmaf(b, yf, c) * inv;
                const int   xi = __float2int_rn(xx);
                if ((unsigned)xi < (unsigned)IMG_W)
                    acc += lds[y * LDS_STRIDE + xi];
                yf += 1.0f;
            }
        }
        outp[v * IMG_W + u] = acc;                        // coalesced b32 store
    }
}

extern "C" void kernel_launch(void* const* d_in, const int* in_sizes, int n_in,
                              void* d_out, int out_size, void* d_ws, size_t ws_size,
                              hipStream_t stream) {
    (void)n_in; (void)out_size; (void)d_ws; (void)ws_size;
    const float* img = (const float*)d_in[0];   // (2, 8, 256, 256) f32
    const float* Fm  = (const float*)d_in[1];   // (3, 3) f32
    float*       out = (float*)d_out;           // (2, 8, 256, 256) f32

    const int nplanes = in_sizes[0] / (IMG_H * IMG_W);   // 16
    dim3 grid(IMG_H / ROWS_PER_TILE, nplanes);           // 16 x 16 = 256 blocks
    epiline_sum_kernel<<<grid, BLOCK_THREADS, 0, stream>>>(img, Fm, out);
}